// LocalLinearLayer_10806137717199
// MI455X (gfx1250) — compile-verified
//
#include <hip/hip_runtime.h>

// f32 WMMA fragments: A/B are 16x4 / 4x16 f32 (2 VGPRs each), C/D is 16x16 f32 (8 VGPRs)
typedef __attribute__((ext_vector_type(2))) float v2f;
typedef __attribute__((ext_vector_type(8))) float v8f;

#define LEN    4096   // sequence length (in == out)
#define CH     64     // channels
#define WIN    25     // window
#define PADV   12     // (WIN-1)/2
#define PADDED 4120   // LEN + 2*PADV
#define TILE   16     // output positions per tile
#define NSTEP  10     // 40-wide contraction window / K=4 per WMMA

// One block = one (batch, 16-position tile). 4 waves, each owns a 16-channel slab.
// EDGE=false: interior tiles (1..254) where the padded index is uniformly l = p - 12,
//             so all loads hang off one hoisted base pointer with immediate offsets.
// EDGE=true:  tiles 0 and 255, full edge-copy remap per position.
template <bool EDGE>
__global__ __launch_bounds__(128)
void local_linear_wmma(const float* __restrict__ x,    // [32, 4096, 64]
                       const float* __restrict__ W,    // [4096, 4120] (only band used)
                       const float* __restrict__ bias, // [4096]
                       float* __restrict__ out,        // [32, 4096, 64]
                       int tile_off, int tile_mul)
{
    const int tile  = tile_off + (int)blockIdx.x * tile_mul;
    const int batch = blockIdx.y;
    const int wave  = threadIdx.x >> 5;    // 0..3 -> channel chunk
    const int lane  = threadIdx.x & 31;
    const int half  = lane >> 4;           // 0: lanes 0-15, 1: lanes 16-31
    const int m     = lane & 15;           // A row / D column index

    const int    t16   = tile * TILE;
    const int    cbase = wave * 16;
    const size_t xbase = (size_t)batch * (LEN * CH);

    // Hoisted per-lane bases. K index this lane holds at step s: j = 4s + 2*half + {0,1}.
    // W row o = t16+m, window column base t16 + 2*half  (even float offset -> b64-aligned).
    const float* wrow  = W + (size_t)(t16 + m) * PADDED + (t16 + 2 * half);
    // Interior x base: position l = t16 - PADV + 2*half, channel cbase + m.
    const float* xlane = x + xbase + (size_t)(t16 - PADV + 2 * half) * CH + (cbase + m);

    v8f acc = {};

#pragma unroll
    for (int s = 0; s < NSTEP; ++s) {
        const int j0 = 4 * s + 2 * half;

        // ---- A fragment: unconditional b64 load of the weight pair, then band mask
        //      applied as a multiply (keeps the load unpredicated; W holds garbage
        //      outside the band j in [m, m+WIN)).
        const v2f  av = *(const v2f*)(wrow + 4 * s);
        const float m0 = ((unsigned)(j0     - m) < (unsigned)WIN) ? 1.0f : 0.0f;
        const float m1 = ((unsigned)(j0 + 1 - m) < (unsigned)WIN) ? 1.0f : 0.0f;
        v2f a;
        a.x = av.x * m0;
        a.y = av.y * m1;

        // ---- B fragment: padded input, N = channel, K = padded position.
        v2f bf;
        if (EDGE) {
            // Edge-copy padding: p<12 -> x[p]; p<4108 -> x[p-12]; else x[p-24].
            const int p0 = t16 + j0;
            const int p1 = p0 + 1;
            const int l0 = (p0 < PADV) ? p0 : ((p0 < LEN + PADV) ? p0 - PADV : p0 - 2 * PADV);
            const int l1 = (p1 < PADV) ? p1 : ((p1 < LEN + PADV) ? p1 - PADV : p1 - 2 * PADV);
            bf.x = x[xbase + (size_t)l0 * CH + cbase + m];
            bf.y = x[xbase + (size_t)l1 * CH + cbase + m];
        } else {
            // Immediate offsets: (4s)*256 B and (4s+1)*256 B off one base.
            bf.x = xlane[(4 * s)     * CH];
            bf.y = xlane[(4 * s + 1) * CH];
        }

        // D = A x B + C, full f32 (bandwidth-bound; no reason to drop precision).
        acc = __builtin_amdgcn_wmma_f32_16x16x4_f32(
            /*neg_a=*/false, a, /*neg_b=*/false, bf,
            /*c_mod=*/(short)0, acc, /*reuse_a=*/false, /*reuse_b=*/false);
    }

    // D layout: VGPR v, lanes 0-15 -> M=v ; lanes 16-31 -> M=v+8 ; N = lane&15.
    // out[b, o, c], o = t16 + v + 8*half, c = cbase + m (16 contiguous floats per store row).
#pragma unroll
    for (int v = 0; v < 8; ++v) {
        const int o = t16 + v + 8 * half;
        out[xbase + (size_t)o * CH + cbase + m] = acc[v] + bias[o];
    }
}

extern "C" void kernel_launch(void* const* d_in, const int* in_sizes, int n_in,
                              void* d_out, int out_size, void* d_ws, size_t ws_size,
                              hipStream_t stream) {
    const float* x   = (const float*)d_in[0]; // [32, 4096, 64]
    const float* W   = (const float*)d_in[1]; // [4096, 4120]
    const float* b   = (const float*)d_in[2]; // [4096]
    float*       out = (float*)d_out;         // [32, 4096, 64]

    dim3 block(128);  // 4 waves x 16 channels

    // Interior tiles 1..254: branch-free fast path.
    local_linear_wmma<false><<<dim3(254, 32), block, 0, stream>>>(x, W, b, out, /*off=*/1, /*mul=*/1);
    // Edge tiles 0 and 255: full padding remap.
    local_linear_wmma<true ><<<dim3(2,   32), block, 0, stream>>>(x, W, b, out, /*off=*/0, /*mul=*/255);
}